// AdaptiveGraphPooling_36034775613468
// MI455X (gfx1250) — compile-verified
//
#include <hip/hip_runtime.h>
#include <math.h>

#define H 128          // feature dim
#define K3H 384        // 3*H
#define WAVES_PER_BLK 8

typedef __attribute__((ext_vector_type(2))) float v2f;
typedef __attribute__((ext_vector_type(8))) float v8f;

__device__ __forceinline__ int lower_bound_i32(const int* __restrict__ b, int n, int v) {
    int lo = 0, hi = n;
    while (lo < hi) {
        int mid = (lo + hi) >> 1;
        if (b[mid] < v) lo = mid + 1; else hi = mid;
    }
    return lo;
}

// One block per graph (batch is sorted). blockDim = 256 (8 waves of 32).
// Wave w handles rows start+w, start+w+8, ...; lane l handles features 4l..4l+3
// (one GLOBAL_LOAD_B128 per wave per row). Single pass over x:
//   - sum / max pooling accumulated per wave in registers
//   - gate = dot(x_row, gate_w)+b via 5 shfl_xor hops (wave32 allreduce)
//   - online softmax (running m, s) with rescaled attention accumulator
//     acc = sum_i exp(gate_i - m) * x_i   (flash-attention style)
// Cross-wave merge through LDS at the end.
__global__ void segment_pool_kernel(const float* __restrict__ x,
                                    const int* __restrict__ batch,
                                    const float* __restrict__ gate_w,
                                    const float* __restrict__ gate_b,
                                    float* __restrict__ combined,  // [G, 384]
                                    int N) {
    const int g    = blockIdx.x;
    const int tid  = threadIdx.x;
    const int wave = tid >> 5;
    const int lane = tid & 31;

    const int start = lower_bound_i32(batch, N, g);
    const int end   = lower_bound_i32(batch, N, g + 1);

    __shared__ float s_sum[WAVES_PER_BLK][H];   // per-wave partial sums
    __shared__ float s_mx [WAVES_PER_BLK][H];   // per-wave partial max
    __shared__ float s_att[WAVES_PER_BLK][H];   // per-wave attention accumulators
    __shared__ float s_m[WAVES_PER_BLK];        // per-wave online-softmax max
    __shared__ float s_s[WAVES_PER_BLK];        // per-wave online-softmax sum

    const float4 gw = reinterpret_cast<const float4*>(gate_w)[lane];
    const float  gb = gate_b[0];

    float sx = 0.f, sy = 0.f, sz = 0.f, sw = 0.f;
    float mx = -INFINITY, my = -INFINITY, mz = -INFINITY, mw = -INFINITY;
    float ax = 0.f, ay = 0.f, az = 0.f, aw = 0.f;    // rescaled attention acc
    float run_m = -INFINITY, run_s = 0.f;

    for (int i = start + wave; i < end; i += WAVES_PER_BLK) {
        const float4 v = reinterpret_cast<const float4*>(x + (size_t)i * H)[lane];
        sx += v.x; sy += v.y; sz += v.z; sw += v.w;
        mx = fmaxf(mx, v.x); my = fmaxf(my, v.y);
        mz = fmaxf(mz, v.z); mw = fmaxf(mw, v.w);

        float d = v.x * gw.x + v.y * gw.y + v.z * gw.z + v.w * gw.w;
        d += __shfl_xor(d, 16); d += __shfl_xor(d, 8);
        d += __shfl_xor(d, 4);  d += __shfl_xor(d, 2);
        d += __shfl_xor(d, 1);
        const float gate  = d + gb;                   // uniform across wave
        const float mn    = fmaxf(run_m, gate);
        const float scale = __expf(run_m - mn);       // 0 on first row (-inf - mn)
        const float e     = __expf(gate - mn);
        run_s = run_s * scale + e;
        ax = ax * scale + e * v.x;  ay = ay * scale + e * v.y;
        az = az * scale + e * v.z;  aw = aw * scale + e * v.w;
        run_m = mn;
    }

    s_sum[wave][4 * lane + 0] = sx; s_sum[wave][4 * lane + 1] = sy;
    s_sum[wave][4 * lane + 2] = sz; s_sum[wave][4 * lane + 3] = sw;
    s_mx [wave][4 * lane + 0] = mx; s_mx [wave][4 * lane + 1] = my;
    s_mx [wave][4 * lane + 2] = mz; s_mx [wave][4 * lane + 3] = mw;
    s_att[wave][4 * lane + 0] = ax; s_att[wave][4 * lane + 1] = ay;
    s_att[wave][4 * lane + 2] = az; s_att[wave][4 * lane + 3] = aw;
    if (lane == 0) { s_m[wave] = run_m; s_s[wave] = run_s; }
    __syncthreads();

    // Merge per-wave softmax stats (uniform across all threads)
    float M = -INFINITY;
    #pragma unroll
    for (int w = 0; w < WAVES_PER_BLK; ++w) M = fmaxf(M, s_m[w]);
    const float gmax = isfinite(M) ? M : 0.f;        // reference's finite guard
    float denom = 0.f;
    float ew[WAVES_PER_BLK];
    #pragma unroll
    for (int w = 0; w < WAVES_PER_BLK; ++w) {
        const float e = __expf(s_m[w] - gmax);       // 0 for empty waves (m=-inf)
        ew[w] = (s_s[w] > 0.f) ? e : 0.f;            // kill 0*exp(-inf) NaN paths
        denom += s_s[w] * ew[w];
    }
    const float inv_denom = 1.f / fmaxf(denom, 1e-16f);

    if (tid < H) {
        float ts = 0.f, tm = -INFINITY, ta = 0.f;
        #pragma unroll
        for (int w = 0; w < WAVES_PER_BLK; ++w) {
            ts += s_sum[w][tid];
            tm  = fmaxf(tm, s_mx[w][tid]);
            ta += s_att[w][tid] * ew[w];
        }
        const float cnt = (float)(end - start);
        float* row = combined + (size_t)g * K3H;
        row[tid]         = ta * inv_denom;               // attention pool
        row[H + tid]     = ts / fmaxf(cnt, 1.f);         // mean pool
        row[2 * H + tid] = isfinite(tm) ? tm : 0.f;      // max pool (finite guard)
    }
}

// out[G,128] = combined[G,384] @ W[384,128] + bias, via V_WMMA_F32_16X16X4_F32.
// One wave per 16x16 output tile; grid sized so EXEC is all-ones (WMMA requirement).
// A-frag (16x4 f32, 2 VGPR): lanes 0-15 hold K={k0,k0+1}, lanes 16-31 K={k0+2,k0+3}.
// B-frag (4x16 f32, 2 VGPR): mirrored; C/D v8f: row = r + 8*(lane>=16), col = lane&15.
__global__ void combine_gemm_kernel(const float* __restrict__ Cmb,   // [G,384]
                                    const float* __restrict__ W,     // [384,128]
                                    const float* __restrict__ bias,  // [128]
                                    float* __restrict__ out) {       // [G,128]
    const int gwave  = (blockIdx.x * blockDim.x + threadIdx.x) >> 5;
    const int lane   = threadIdx.x & 31;
    const int tile_n = gwave & 7;          // 128/16 = 8 column tiles
    const int tile_m = gwave >> 3;
    const int m0 = tile_m * 16;
    const int n0 = tile_n * 16;
    const int half = lane >> 4;            // 0: lanes 0-15, 1: lanes 16-31
    const int lid  = lane & 15;

    const float* arow = Cmb + (size_t)(m0 + lid) * K3H + 2 * half;
    const float* bcol = W + (size_t)(2 * half) * H + n0 + lid;

    v8f c = {};
    for (int k0 = 0; k0 < K3H; k0 += 4) {
        v2f a;
        a.x = arow[k0];
        a.y = arow[k0 + 1];
        v2f b;
        b.x = bcol[(size_t)k0 * H];
        b.y = bcol[(size_t)(k0 + 1) * H];
        c = __builtin_amdgcn_wmma_f32_16x16x4_f32(false, a, false, b,
                                                  (short)0, c, false, false);
    }

    const float bv = bias[n0 + lid];
    #pragma unroll
    for (int r = 0; r < 8; ++r) {
        const int row = m0 + r + 8 * half;
        out[(size_t)row * H + n0 + lid] = c[r] + bv;
    }
}

extern "C" void kernel_launch(void* const* d_in, const int* in_sizes, int n_in,
                              void* d_out, int out_size, void* d_ws, size_t ws_size,
                              hipStream_t stream) {
    const float* x         = (const float*)d_in[0];
    const int*   batch     = (const int*)  d_in[1];
    const float* gate_w    = (const float*)d_in[2];
    const float* gate_b    = (const float*)d_in[3];
    const float* combine_w = (const float*)d_in[4];
    const float* combine_b = (const float*)d_in[5];

    const int N = in_sizes[0] / H;       // 500000
    const int G = out_size   / H;        // 1024

    float* combined = (float*)d_ws;      // G * 384 floats of scratch

    segment_pool_kernel<<<G, 256, 0, stream>>>(x, batch, gate_w, gate_b, combined, N);

    // (G/16)*(128/16) = 512 tiles -> 512 waves -> 8 waves/block -> G/16 blocks
    combine_gemm_kernel<<<G / 16, 256, 0, stream>>>(combined, combine_w, combine_b,
                                                    (float*)d_out);
}